// SoftmaxAttention_24687472017545
// MI455X (gfx1250) — compile-verified
//
#include <hip/hip_runtime.h>

typedef float v2f __attribute__((ext_vector_type(2)));
typedef float v4f __attribute__((ext_vector_type(4)));
typedef float v8f __attribute__((ext_vector_type(8)));

#define B_SZ   32
#define N_SZ   4096
#define DIN    1024

// ---------------------------------------------------------------------------
// Kernel 1: logits[r] = dot(x[r, :], w_x)   for r in [0, B*N)
// Wave-level: D(16x16) = A(16x4:x rows) * B(4x16:w broadcast) accumulated over K.
// Column N=0 of D carries the 16 row dot-products.
// ---------------------------------------------------------------------------
__global__ __launch_bounds__(256) void gemv_logits_wmma(
    const float* __restrict__ x, const float* __restrict__ w,
    float* __restrict__ logits)
{
    __shared__ __attribute__((aligned(16))) float wlds[DIN];
    const int t = threadIdx.x;
    // stage w_x (4 KB) into LDS: 256 threads x float4 = 1024 floats
    ((v4f*)wlds)[t] = ((const v4f*)w)[t];
    __syncthreads();

    const int lane = t & 31;
    const int wave = t >> 5;
    const size_t tile = (size_t)blockIdx.x * 128u + (size_t)wave * 16u;
    const float* xr = x + (tile + (size_t)(lane & 15)) * DIN;
    const int koff = (lane >> 4) << 1;   // lanes 0-15 -> K 0,1 ; lanes 16-31 -> K 2,3

    v8f c0 = {};
    v8f c1 = {};
    #pragma unroll 8
    for (int kk = 0; kk < DIN; kk += 8) {
        // A-matrix fragment: (M = lane%16, K = koff..koff+1)
        v2f a0 = *(const v2f*)(xr + kk + koff);
        v2f a1 = *(const v2f*)(xr + kk + 4 + koff);
        // B-matrix fragment: w broadcast across all 16 columns
        v2f b0 = *(const v2f*)(wlds + kk + koff);
        v2f b1 = *(const v2f*)(wlds + kk + 4 + koff);
        c0 = __builtin_amdgcn_wmma_f32_16x16x4_f32(false, a0, false, b0,
                                                   (short)0, c0, false, false);
        c1 = __builtin_amdgcn_wmma_f32_16x16x4_f32(false, a1, false, b1,
                                                   (short)0, c1, false, false);
    }

    // D column N==0: lane 0 has rows tile+0..7 (c[0..7]), lane 16 has rows tile+8..15
    if ((lane & 15) == 0) {
        const size_t base = tile + (size_t)((lane >> 4) << 3);
        #pragma unroll
        for (int v = 0; v < 8; ++v) logits[base + v] = c0[v] + c1[v];
    }
}

// ---------------------------------------------------------------------------
// Kernel 2: out[b, :] = softmax(logits[b, :] + hidden[b]@w_h + bias)
// One 1024-thread block per batch row (4096 elements -> 4 per thread).
// ---------------------------------------------------------------------------
__device__ __forceinline__ float wred_max(float v) {
    #pragma unroll
    for (int o = 16; o > 0; o >>= 1) v = fmaxf(v, __shfl_xor(v, o, 32));
    return v;
}
__device__ __forceinline__ float wred_sum(float v) {
    #pragma unroll
    for (int o = 16; o > 0; o >>= 1) v += __shfl_xor(v, o, 32);
    return v;
}

__global__ __launch_bounds__(1024) void softmax_rows(
    const float* __restrict__ logits, const float* __restrict__ hidden,
    const float* __restrict__ w_h, const float* __restrict__ bias,
    float* __restrict__ out)
{
    __shared__ float red[32];
    const int b    = blockIdx.x;
    const int t    = threadIdx.x;
    const int lane = t & 31;
    const int wv   = t >> 5;
    const float* row  = logits + (size_t)b * N_SZ;
    float*       orow = out    + (size_t)b * N_SZ;

    // hidden term, constant over n: hb = dot(hidden[b], w_h) + bias
    float ph = hidden[(size_t)b * 1024 + t] * w_h[t];
    ph = wred_sum(ph);
    if (lane == 0) red[wv] = ph;
    __syncthreads();
    float hb = wred_sum(red[lane]) + bias[0];
    __syncthreads();

    float v[4];
    float m = -3.402823466e38f;
    #pragma unroll
    for (int i = 0; i < 4; ++i) {
        v[i] = row[t + i * 1024] + hb;
        m = fmaxf(m, v[i]);
    }
    m = wred_max(m);
    if (lane == 0) red[wv] = m;
    __syncthreads();
    m = wred_max(red[lane]);
    __syncthreads();

    float s = 0.0f;
    #pragma unroll
    for (int i = 0; i < 4; ++i) {
        v[i] = __expf(v[i] - m);
        s += v[i];
    }
    s = wred_sum(s);
    if (lane == 0) red[wv] = s;
    __syncthreads();
    s = wred_sum(red[lane]);

    const float inv = 1.0f / s;
    #pragma unroll
    for (int i = 0; i < 4; ++i) orow[t + i * 1024] = v[i] * inv;
}

// ---------------------------------------------------------------------------
extern "C" void kernel_launch(void* const* d_in, const int* in_sizes, int n_in,
                              void* d_out, int out_size, void* d_ws, size_t ws_size,
                              hipStream_t stream) {
    const float* x      = (const float*)d_in[0];  // [B, N, D_IN]
    const float* hidden = (const float*)d_in[1];  // [B, D_STATE]
    const float* w_x    = (const float*)d_in[2];  // [D_IN]
    const float* w_h    = (const float*)d_in[3];  // [D_STATE]
    const float* bias   = (const float*)d_in[4];  // scalar
    float* out    = (float*)d_out;                // [B, N]
    float* logits = (float*)d_ws;                 // B*N floats = 512 KiB scratch

    dim3 grid1((B_SZ * N_SZ) / 128);              // 8 waves/WG * 16 rows/wave
    gemv_logits_wmma<<<grid1, 256, 0, stream>>>(x, w_x, logits);
    softmax_rows<<<B_SZ, 1024, 0, stream>>>(logits, hidden, w_h, bias, out);
}